// DigitCaps_33689723470310
// MI455X (gfx1250) — compile-verified
//
#include <hip/hip_runtime.h>
#include <hip/hip_bf16.h>
#include <math.h>

// ---------------------------------------------------------------------------
// DigitCaps routing (CapsNet) for MI455X / gfx1250, wave32, WMMA f32 16x16x4.
//
// Algebraic reduction: A_sum[b,d,m] = T[b,d,:] . U_hat[b,d,m,:] / sqrt(dp)
// with T = sum_n U_hat, so the NxN similarity matrix is never materialized.
// S[b,d,j] = SB[b,d,j] (prior-weighted, pass1) + S2[b,d,j] (softmax-weighted,
// pass3). U_hat tiles are recomputed per pass via V_WMMA_F32_16X16X4_F32
// (two chained K=4 ops give the K=8 contraction); W (5.9MB) lives in L2.
// ---------------------------------------------------------------------------

typedef float v2f __attribute__((ext_vector_type(2)));
typedef float v8f __attribute__((ext_vector_type(8)));

#define NN 1152
#define DD 10
#define BB 16

// D-tile layout (32-bit C/D 16x16): lane l, vgpr v  ->  (b = v + 8*(l>=16), j = l&15)

__device__ __forceinline__ v8f uhat_tile(const float* __restrict__ up,
                                         const float* __restrict__ wp) {
  // up -> &u[b][n][2*hi],  wp -> &W[d][n][j][2*hi]   (b = j = lane&15, hi = lane>>4)
  v2f a0 = *(const v2f*)(up);       // K = {0,1} (lo lanes) / {2,3} (hi lanes)
  v2f a1 = *(const v2f*)(up + 4);   // K' for second wmma (i = 4..7)
  v2f b0 = *(const v2f*)(wp);
  v2f b1 = *(const v2f*)(wp + 4);
  v8f acc = {0.f,0.f,0.f,0.f,0.f,0.f,0.f,0.f};
  acc = __builtin_amdgcn_wmma_f32_16x16x4_f32(false, a0, false, b0,
                                              (short)0, acc, false, false);
  acc = __builtin_amdgcn_wmma_f32_16x16x4_f32(false, a1, false, b1,
                                              (short)0, acc, false, false);
  return acc;
}

__global__ __launch_bounds__(256) void caps_init(float* __restrict__ ws, int n) {
  int i = blockIdx.x * 256 + threadIdx.x;
  if (i < n) ws[i] = 0.f;
}

// Pass 1: T[d][b][j] = sum_n U_hat ;  SB[d][b][j] = sum_n bp[d,n]*U_hat
__global__ __launch_bounds__(256) void caps_pass1(
    const float* __restrict__ u,   // [16,1152,8]
    const float* __restrict__ W,   // [10,1152,16,8]
    const float* __restrict__ bp,  // [10,1,1152]
    float* __restrict__ T,         // [10,16,16]
    float* __restrict__ SB) {      // [10,16,16]
  const int d    = blockIdx.x;
  const int wave = threadIdx.x >> 5;
  const int lane = threadIdx.x & 31;
  const int hi   = lane >> 4;
  const int col  = lane & 15;            // b for A-frag, j for B-frag
  const int n0   = blockIdx.y * 64 + wave * 8;

  const float* uB = u + (size_t)col * (NN * 8) + 2 * hi;
  const float* wB = W + ((size_t)d * NN * 16 + col) * 8 + 2 * hi;

  v8f accT = {0.f,0.f,0.f,0.f,0.f,0.f,0.f,0.f};
  v8f accS = accT;
  for (int k = 0; k < 8; ++k) {
    const int n = n0 + k;
    v8f uh = uhat_tile(uB + (size_t)n * 8, wB + (size_t)n * 128);
    const float p = bp[d * NN + n];
#pragma unroll
    for (int v = 0; v < 8; ++v) { accT[v] += uh[v]; accS[v] += p * uh[v]; }
  }
#pragma unroll
  for (int v = 0; v < 8; ++v) {
    const int b = v + 8 * hi;
    atomicAdd(&T [(d * 16 + b) * 16 + col], accT[v]);
    atomicAdd(&SB[(d * 16 + b) * 16 + col], accS[v]);
  }
}

// Pass 2: Asum[d][n][b] = (T[d][b][:] . U_hat[b,d,n,:]) / sqrt(8)
__global__ __launch_bounds__(256) void caps_pass2(
    const float* __restrict__ u, const float* __restrict__ W,
    const float* __restrict__ T, float* __restrict__ Asum) {
  const int d    = blockIdx.x;
  const int wave = threadIdx.x >> 5;
  const int lane = threadIdx.x & 31;
  const int hi   = lane >> 4;
  const int col  = lane & 15;
  const int n0   = blockIdx.y * 64 + wave * 8;

  const float* uB = u + (size_t)col * (NN * 8) + 2 * hi;
  const float* wB = W + ((size_t)d * NN * 16 + col) * 8 + 2 * hi;

  float tv[8];
#pragma unroll
  for (int v = 0; v < 8; ++v) tv[v] = T[(d * 16 + v + 8 * hi) * 16 + col];

  const float rsq = 0.35355339059327373f;  // 1/sqrt(8)
  for (int k = 0; k < 8; ++k) {
    const int n = n0 + k;
    v8f uh = uhat_tile(uB + (size_t)n * 8, wB + (size_t)n * 128);
#pragma unroll
    for (int v = 0; v < 8; ++v) {
      float p = uh[v] * tv[v];                 // term (b = v+8hi, j = col)
      p += __shfl_xor(p, 1, 32);               // reduce over j (16 lanes per half)
      p += __shfl_xor(p, 2, 32);
      p += __shfl_xor(p, 4, 32);
      p += __shfl_xor(p, 8, 32);
      if (col == 0) Asum[((size_t)d * NN + n) * 16 + v + 8 * hi] = p * rsq;
    }
  }
}

// Softmax over d for each (b,n):  C[d][n][b]
__global__ __launch_bounds__(256) void caps_softmax(
    const float* __restrict__ Asum, float* __restrict__ C) {
  const int idx = blockIdx.x * 256 + threadIdx.x;
  if (idx >= BB * NN) return;
  const int b = idx & 15, n = idx >> 4;
  float a[DD], m = -3.402823466e+38f;
#pragma unroll
  for (int d = 0; d < DD; ++d) {
    a[d] = Asum[((size_t)d * NN + n) * 16 + b];
    m = fmaxf(m, a[d]);
  }
  float s = 0.f;
#pragma unroll
  for (int d = 0; d < DD; ++d) { a[d] = __expf(a[d] - m); s += a[d]; }
  const float inv = 1.f / s;
#pragma unroll
  for (int d = 0; d < DD; ++d) C[((size_t)d * NN + n) * 16 + b] = a[d] * inv;
}

// Pass 3: S2[d][b][j] = sum_n C[b,d,n] * U_hat[b,d,n,j]
__global__ __launch_bounds__(256) void caps_pass3(
    const float* __restrict__ u, const float* __restrict__ W,
    const float* __restrict__ C, float* __restrict__ S2) {
  const int d    = blockIdx.x;
  const int wave = threadIdx.x >> 5;
  const int lane = threadIdx.x & 31;
  const int hi   = lane >> 4;
  const int col  = lane & 15;
  const int n0   = blockIdx.y * 64 + wave * 8;

  const float* uB = u + (size_t)col * (NN * 8) + 2 * hi;
  const float* wB = W + ((size_t)d * NN * 16 + col) * 8 + 2 * hi;

  v8f acc = {0.f,0.f,0.f,0.f,0.f,0.f,0.f,0.f};
  for (int k = 0; k < 8; ++k) {
    const int n = n0 + k;
    v8f uh = uhat_tile(uB + (size_t)n * 8, wB + (size_t)n * 128);
    const float4* cp = (const float4*)(C + ((size_t)d * NN + n) * 16 + 8 * hi);
    const float4 c0 = cp[0], c1 = cp[1];     // C for b = 8*hi + {0..3}, {4..7}
    acc[0] += c0.x * uh[0]; acc[1] += c0.y * uh[1];
    acc[2] += c0.z * uh[2]; acc[3] += c0.w * uh[3];
    acc[4] += c1.x * uh[4]; acc[5] += c1.y * uh[5];
    acc[6] += c1.z * uh[6]; acc[7] += c1.w * uh[7];
  }
#pragma unroll
  for (int v = 0; v < 8; ++v)
    atomicAdd(&S2[(d * 16 + v + 8 * hi) * 16 + col], acc[v]);
}

// Final: S = SB + S2 ; squash ; write out[b][d][j]
__global__ __launch_bounds__(192) void caps_final(
    const float* __restrict__ SB, const float* __restrict__ S2,
    float* __restrict__ out) {
  const int t = threadIdx.x;
  if (t >= BB * DD) return;
  const int b = t / DD, d = t % DD;
  float s[16], nrm2 = 0.f;
#pragma unroll
  for (int j = 0; j < 16; ++j) {
    s[j] = SB[(d * 16 + b) * 16 + j] + S2[(d * 16 + b) * 16 + j];
    nrm2 += s[j] * s[j];
  }
  const float nrm  = sqrtf(nrm2);
  const float coef = 1.f - 1.f / (__expf(nrm) + 1e-7f);
  const float sc   = coef / (nrm + 1e-7f);
#pragma unroll
  for (int j = 0; j < 16; ++j) out[(b * DD + d) * 16 + j] = sc * s[j];
}

extern "C" void kernel_launch(void* const* d_in, const int* in_sizes, int n_in,
                              void* d_out, int out_size, void* d_ws, size_t ws_size,
                              hipStream_t stream) {
  const float* u  = (const float*)d_in[0];   // [16,1152,8]
  const float* W  = (const float*)d_in[1];   // [10,1152,16,8]
  const float* bp = (const float*)d_in[2];   // [10,1,1152]
  float* out = (float*)d_out;                // [16,10,16]

  float* wsf  = (float*)d_ws;
  float* T    = wsf;              // 2560 floats
  float* SB   = wsf + 2560;       // 2560
  float* S2   = wsf + 5120;       // 2560
  float* Asum = wsf + 7680;       // 10*1152*16 = 184320
  float* Cm   = wsf + 192000;     // 184320   (total ~1.47 MB)

  caps_init<<<(7680 + 255) / 256, 256, 0, stream>>>(wsf, 7680);

  dim3 grid(DD, 18);              // 10 digit caps x 18 chunks of 64 n
  caps_pass1  <<<grid, 256, 0, stream>>>(u, W, bp, T, SB);
  caps_pass2  <<<grid, 256, 0, stream>>>(u, W, T, Asum);
  caps_softmax<<<(BB * NN + 255) / 256, 256, 0, stream>>>(Asum, Cm);
  caps_pass3  <<<grid, 256, 0, stream>>>(u, W, Cm, S2);
  caps_final  <<<1, 192, 0, stream>>>(SB, S2, out);
}